// GATConv_9328668967305
// MI455X (gfx1250) — compile-verified
//
#include <hip/hip_runtime.h>
#include <hip/hip_bf16.h>

// ---------------------------------------------------------------------------
// GATConv on MI455X (gfx1250, wave32)
//   out = segment_softmax_scatter( x@W1^T , attention(s_i, s_j) )
// s_i = x @ (W2^T a_s), s_j = x @ (W1^T a_t)  -- second GEMM collapsed to matvec.
// GEMM uses V_WMMA_F32_16X16X4_F32 (full fp32 precision, CDNA5 matrix pipe).
// x-tile staging uses GLOBAL_LOAD_ASYNC_TO_LDS_B128 (ASYNCcnt) when available.
// ---------------------------------------------------------------------------

typedef float  v2f __attribute__((ext_vector_type(2)));
typedef float  v8f __attribute__((ext_vector_type(8)));

#define NEG_SLOPE 0.2f
#define D 128          // feature dim (== H)
#define LDS_STRIDE 132 // 128 + 4 pad: rows land on distinct banks

#if defined(__has_builtin)
#  if __has_builtin(__builtin_amdgcn_global_load_async_to_lds_b128)
#    define HAVE_ASYNC_LDS 1
#  endif
#endif
#ifndef HAVE_ASYNC_LDS
#  define HAVE_ASYNC_LDS 0
#endif

// Builtin signature (from hipcc diagnostic): param 1 is
// 'int __attribute__((vector_size(16))) __device__ *' i.e. AS1 v4i*, then LDS ptr.
typedef int v4i_vec __attribute__((vector_size(16)));
typedef __attribute__((address_space(1))) v4i_vec* gptr_b128;
typedef __attribute__((address_space(3))) v4i_vec* lptr_b128;

__device__ __forceinline__ void wait_asynccnt0() {
#if defined(__has_builtin) && __has_builtin(__builtin_amdgcn_s_wait_asynccnt)
    __builtin_amdgcn_s_wait_asynccnt(0);
#else
    asm volatile("s_wait_asynccnt 0x0" ::: "memory");
#endif
}

__device__ __forceinline__ float lrelu(float v) { return v > 0.0f ? v : NEG_SLOPE * v; }

// Monotonic int encoding of IEEE float (involution) -> atomicMax(int) == float max.
__device__ __forceinline__ int   ord_f32(float f) { int i = __float_as_int(f); return i ^ ((i >> 31) & 0x7fffffff); }
__device__ __forceinline__ float unord_f32(int o) { int i = o ^ ((o >> 31) & 0x7fffffff); return __int_as_float(i); }

// ---------------------------------------------------------------------------
// 0) init: out = 0 (float4 stores), denom = 0, m_ord = INT_MIN (ordered -inf)
// ---------------------------------------------------------------------------
__global__ void gat_init_kernel(float4* __restrict__ out4, float* __restrict__ denom,
                                int* __restrict__ m_ord, int N) {
    const int stride = gridDim.x * blockDim.x;
    const int total4 = N * (D / 4);
    const float4 z = make_float4(0.f, 0.f, 0.f, 0.f);
    for (int i = blockIdx.x * blockDim.x + threadIdx.x; i < total4; i += stride) out4[i] = z;
    for (int i = blockIdx.x * blockDim.x + threadIdx.x; i < N; i += stride) {
        denom[i] = 0.0f;
        m_ord[i] = (int)0x80000000; // <= ord(-inf)
    }
}

// ---------------------------------------------------------------------------
// 1) v1 = W1^T a_t,  v2 = W2^T a_s   (128x128 matvecs, one block)
// ---------------------------------------------------------------------------
__global__ void gat_matvec_kernel(const float* __restrict__ W1, const float* __restrict__ W2,
                                  const float* __restrict__ a_s, const float* __restrict__ a_t,
                                  float* __restrict__ v1, float* __restrict__ v2) {
    const int d = threadIdx.x; // 128 threads
    float acc1 = 0.0f, acc2 = 0.0f;
    for (int h = 0; h < D; ++h) {
        acc1 += a_t[h] * W1[h * D + d];
        acc2 += a_s[h] * W2[h * D + d];
    }
    v1[d] = acc1;
    v2[d] = acc2;
}

// ---------------------------------------------------------------------------
// 2) WMMA GEMM: wx[n][h] = sum_d x[n][d] * W1[h][d]   (+ fused s_i/s_j dots)
//    Block = 256 threads (8 waves). Block owns a 16-row tile of x (in LDS,
//    staged via async-to-LDS when available); wave w computes output columns
//    [16w, 16w+16) with 32 K-steps of v_wmma_f32_16x16x4_f32.
// A frag (16x4, MxK): lanes 0-15 row M=lane hold {K0,K1}; lanes 16-31 hold {K2,K3}.
// B frag (4x16, KxN): lanes 0-15 col N=lane hold {K0,K1}; lanes 16-31 hold {K2,K3}.
//                     B[k][n] = W1[col][k] -> consecutive floats of W1 row.
// D frag (16x16): VGPR r -> M=r (lanes 0-15) / M=r+8 (lanes 16-31), N = lane&15.
// ---------------------------------------------------------------------------
__global__ void __launch_bounds__(256)
gat_gemm_s_kernel(const float* __restrict__ x, const float* __restrict__ W1,
                  const float* __restrict__ v1, const float* __restrict__ v2,
                  float* __restrict__ wx, float* __restrict__ s_i, float* __restrict__ s_j,
                  int N) {
    __shared__ float xt[16 * LDS_STRIDE];

    const int tid = threadIdx.x;
    const int m0  = blockIdx.x * 16;
    const bool full_tile = (m0 + 16 <= N); // block-uniform

    // ---- stage 16x128 x-tile into LDS (8 floats per thread) ----
    {
        const int r = tid >> 4;        // 0..15
        const int c = (tid & 15) * 8;  // 0,8,...,120
        const int row = m0 + r;
#if HAVE_ASYNC_LDS
        if (full_tile) {
            // CDNA5 async copy: global -> LDS without VGPR round-trip (ASYNCcnt).
            float* gsrc = const_cast<float*>(x + (size_t)row * D + c);
            float* ldst = &xt[r * LDS_STRIDE + c];
            __builtin_amdgcn_global_load_async_to_lds_b128(
                (gptr_b128)gsrc, (lptr_b128)ldst, 0, 0);
            __builtin_amdgcn_global_load_async_to_lds_b128(
                (gptr_b128)(gsrc + 4), (lptr_b128)(ldst + 4), 0, 0);
            wait_asynccnt0();
        } else
#endif
        {
            const int crow = row < N ? row : N - 1;
            const float4* src = (const float4*)(x + (size_t)crow * D + c);
            float4 p0 = src[0];
            float4 p1 = src[1];
            if (row >= N) { p0 = make_float4(0.f, 0.f, 0.f, 0.f); p1 = p0; }
            *(float4*)&xt[r * LDS_STRIDE + c]     = p0;
            *(float4*)&xt[r * LDS_STRIDE + c + 4] = p1;
        }
    }
    __syncthreads();

    // ---- fused attention-score halves: s_j = x.v1, s_i = x.v2 ----
    {
        const int r = tid >> 4, sub = tid & 15;
        float acc1 = 0.0f, acc2 = 0.0f;
#pragma unroll
        for (int j = 0; j < 8; ++j) {
            const float xv = xt[r * LDS_STRIDE + sub * 8 + j];
            acc1 += xv * v1[sub * 8 + j];
            acc2 += xv * v2[sub * 8 + j];
        }
#pragma unroll
        for (int m = 8; m >= 1; m >>= 1) {
            acc1 += __shfl_xor(acc1, m, 32);
            acc2 += __shfl_xor(acc2, m, 32);
        }
        const int row = m0 + r;
        if (sub == 0 && row < N) { s_j[row] = acc1; s_i[row] = acc2; }
    }

    // ---- WMMA main loop (uniform control flow: EXEC all ones) ----
    const int wave  = tid >> 5;       // 0..7 -> output column tile
    const int lane  = tid & 31;
    const int lhalf = lane >> 4;      // 0: K0/K1 half, 1: K2/K3 half
    const int l15   = lane & 15;
    const int col0  = wave * 16;

    const float* wrow = W1 + (size_t)(col0 + l15) * D + lhalf * 2; // B source
    const float* arow = &xt[l15 * LDS_STRIDE + lhalf * 2];         // A source

    v8f acc = {};
#pragma unroll 4
    for (int k0 = 0; k0 < D; k0 += 4) {
        const float2 av = *(const float2*)(arow + k0);
        const float2 bv = *(const float2*)(wrow + k0);
        v2f a; a.x = av.x; a.y = av.y;
        v2f b; b.x = bv.x; b.y = bv.y;
        acc = __builtin_amdgcn_wmma_f32_16x16x4_f32(
            /*neg_a=*/false, a, /*neg_b=*/false, b,
            /*c_mod=*/(short)0, acc, /*reuse_a=*/false, /*reuse_b=*/false);
    }

    // ---- epilogue: uniform guard, single base, immediate-offset stores ----
    const int rowbase = m0 + lhalf * 8;
    float* dst = wx + (size_t)rowbase * D + (col0 + l15);
    if (full_tile) {
#pragma unroll
        for (int r = 0; r < 8; ++r) dst[r * D] = acc[r]; // 8 clause-able stores
    } else {
#pragma unroll
        for (int r = 0; r < 8; ++r)
            if (rowbase + r < N) dst[r * D] = acc[r];
    }
}

// ---------------------------------------------------------------------------
// 3) segment max:  m[tgt] = max(m[tgt], leaky_relu(s_i[tgt]+s_j[src]))
//    edges idx<E come from edge_index; idx>=E are self loops.
// ---------------------------------------------------------------------------
__global__ void gat_edge_max_kernel(const int* __restrict__ ei, int E, int N,
                                    const float* __restrict__ s_i, const float* __restrict__ s_j,
                                    int* __restrict__ m_ord) {
    const int M = E + N;
    const int stride = gridDim.x * blockDim.x;
    for (int idx = blockIdx.x * blockDim.x + threadIdx.x; idx < M; idx += stride) {
        const int src = idx < E ? ei[idx]     : idx - E;
        const int tgt = idx < E ? ei[E + idx] : idx - E;
        const float e = lrelu(s_i[tgt] + s_j[src]);
        atomicMax(&m_ord[tgt], ord_f32(e));
    }
}

// ---------------------------------------------------------------------------
// 4) segment sum of exp(e - m[tgt])
// ---------------------------------------------------------------------------
__global__ void gat_edge_sum_kernel(const int* __restrict__ ei, int E, int N,
                                    const float* __restrict__ s_i, const float* __restrict__ s_j,
                                    const int* __restrict__ m_ord, float* __restrict__ denom) {
    const int M = E + N;
    const int stride = gridDim.x * blockDim.x;
    for (int idx = blockIdx.x * blockDim.x + threadIdx.x; idx < M; idx += stride) {
        const int src = idx < E ? ei[idx]     : idx - E;
        const int tgt = idx < E ? ei[E + idx] : idx - E;
        const float e  = lrelu(s_i[tgt] + s_j[src]);
        const float ex = __expf(e - unord_f32(m_ord[tgt]));
        unsafeAtomicAdd(&denom[tgt], ex);
    }
}

// ---------------------------------------------------------------------------
// 5) weighted scatter: out[tgt] += alpha * wx[src]
//    One wave32 per edge; lane covers 4 consecutive columns (float4 gather,
//    4x global_atomic_add_f32). wx stays hot in the 192MB L2.
// ---------------------------------------------------------------------------
__global__ void gat_scatter_kernel(const int* __restrict__ ei, int E, int N,
                                   const float* __restrict__ s_i, const float* __restrict__ s_j,
                                   const int* __restrict__ m_ord, const float* __restrict__ denom,
                                   const float* __restrict__ wx, float* __restrict__ out) {
    const int gtid = blockIdx.x * blockDim.x + threadIdx.x;
    const int lane = gtid & 31;
    const int wid  = gtid >> 5;
    const int nw   = (gridDim.x * blockDim.x) >> 5;
    const int M = E + N;
    for (int idx = wid; idx < M; idx += nw) {
        const int src = idx < E ? ei[idx]     : idx - E;
        const int tgt = idx < E ? ei[E + idx] : idx - E;
        const float e     = lrelu(s_i[tgt] + s_j[src]);
        const float alpha = __expf(e - unord_f32(m_ord[tgt])) / denom[tgt];
        const float4 w = *(const float4*)(wx + (size_t)src * D + lane * 4);
        float* o = out + (size_t)tgt * D + lane * 4;
        unsafeAtomicAdd(o + 0, w.x * alpha);
        unsafeAtomicAdd(o + 1, w.y * alpha);
        unsafeAtomicAdd(o + 2, w.z * alpha);
        unsafeAtomicAdd(o + 3, w.w * alpha);
    }
}

// ---------------------------------------------------------------------------
// host launcher
// ---------------------------------------------------------------------------
extern "C" void kernel_launch(void* const* d_in, const int* in_sizes, int n_in,
                              void* d_out, int out_size, void* d_ws, size_t ws_size,
                              hipStream_t stream) {
    const float* x   = (const float*)d_in[0];
    const int*   ei  = (const int*)  d_in[1];
    const float* W1  = (const float*)d_in[2];
    const float* W2  = (const float*)d_in[3];
    const float* a_s = (const float*)d_in[4];
    const float* a_t = (const float*)d_in[5];

    const int N = in_sizes[0] / D;
    const int E = in_sizes[1] / 2;
    float* out = (float*)d_out;

    // workspace carve-up (256B aligned)
    char*  ws  = (char*)d_ws;
    size_t off = 0;
    auto carve = [&](size_t bytes) -> void* {
        void* p = ws + off;
        off = (off + bytes + 255) & ~(size_t)255;
        return p;
    };
    float* wx    = (float*)carve((size_t)N * D * sizeof(float));
    float* s_i   = (float*)carve((size_t)N * sizeof(float));
    float* s_j   = (float*)carve((size_t)N * sizeof(float));
    int*   m_ord = (int*)  carve((size_t)N * sizeof(int));
    float* denom = (float*)carve((size_t)N * sizeof(float));
    float* v1    = (float*)carve(D * sizeof(float));
    float* v2    = (float*)carve(D * sizeof(float));

    const int M = E + N;

    gat_init_kernel<<<2048, 256, 0, stream>>>((float4*)out, denom, m_ord, N);
    gat_matvec_kernel<<<1, D, 0, stream>>>(W1, W2, a_s, a_t, v1, v2);
    gat_gemm_s_kernel<<<(N + 15) / 16, 256, 0, stream>>>(x, W1, v1, v2, wx, s_i, s_j, N);
    gat_edge_max_kernel<<<(M + 255) / 256, 256, 0, stream>>>(ei, E, N, s_i, s_j, m_ord);
    gat_edge_sum_kernel<<<(M + 255) / 256, 256, 0, stream>>>(ei, E, N, s_i, s_j, m_ord, denom);
    gat_scatter_kernel<<<(M + 7) / 8, 256, 0, stream>>>(ei, E, N, s_i, s_j, m_ord, denom, wx, out);
}